// TwoWordProbe_2508260901508
// MI455X (gfx1250) — compile-verified
//
#include <hip/hip_runtime.h>

typedef float v2f __attribute__((ext_vector_type(2)));
typedef float v8f __attribute__((ext_vector_type(8)));

#define BATCHES   16
#define SEQ       256
#define DIM       1024
#define RANK      128
#define KCHUNK    64
#define LDSA_STR  68      // 16-row A tile, pad 64->68: float4-aligned, 4-bank row spread
#define LDST_STR  132     // 256-row T tile, pad 128->132: float4-aligned, 4-bank row spread

// ---------------------------------------------------------------------------
// Kernel 1: T[b] = batch[b] (256x1024) @ B (1024x128), fp32 WMMA 16x16x4.
// grid = (16 batches, 16 M-tiles), block = 256 threads (8 waves, one N-tile each)
// ---------------------------------------------------------------------------
__global__ __launch_bounds__(256) void proj_wmma(const float* __restrict__ batch,
                                                 const float* __restrict__ Bm,
                                                 float* __restrict__ T) {
  __shared__ float ldsA[16 * LDSA_STR];   // 16 x 64 (padded)
  __shared__ float ldsB[KCHUNK * RANK];   // 64 x 128 (contiguous in B -> no pad needed)

  const int b    = blockIdx.x;
  const int mt   = blockIdx.y;
  const int t    = threadIdx.x;
  const int wave = t >> 5;                // 0..7
  const int lane = t & 31;
  const int lcol = lane & 15;             // M (A-frag) / N (B-frag)
  const int khalf = (lane >> 4) << 1;     // 0 or 2
  const int n0   = wave * 16;

  const float* Ag = batch + (size_t)b * SEQ * DIM + (size_t)(mt * 16) * DIM;

  v8f acc = {};

  for (int k0 = 0; k0 < DIM; k0 += KCHUNK) {
    // ---- stage A chunk: 16 rows x 64 cols, 1024 floats, one float4 per thread
    {
      const int idx = t * 4;              // 0..1023
      const int row = idx >> 6;           // /64
      const int col = idx & 63;
      const float4 v = *reinterpret_cast<const float4*>(Ag + row * DIM + k0 + col);
      *reinterpret_cast<float4*>(&ldsA[row * LDSA_STR + col]) = v;  // 16B aligned (272*row)
    }
    // ---- stage B chunk: rows k0..k0+63 of B are contiguous (64*128 floats)
    {
      const float* Bg = Bm + (size_t)k0 * RANK;
#pragma unroll
      for (int c = 0; c < 8; ++c) {
        const int idx = (c * 256 + t) * 4;
        *reinterpret_cast<float4*>(&ldsB[idx]) = *reinterpret_cast<const float4*>(Bg + idx);
      }
    }
    __syncthreads();

#pragma unroll
    for (int kk = 0; kk < KCHUNK; kk += 4) {
      // A frag 16x4: lane holds A[lcol][kk+khalf], A[lcol][kk+khalf+1]  (ds_load_b64)
      v2f afrag = *reinterpret_cast<const v2f*>(&ldsA[lcol * LDSA_STR + kk + khalf]);
      // B frag 4x16: lane holds B[kk+khalf][n0+lcol], B[kk+khalf+1][n0+lcol]
      v2f bfrag;
      bfrag.x = ldsB[(kk + khalf)     * RANK + n0 + lcol];
      bfrag.y = ldsB[(kk + khalf + 1) * RANK + n0 + lcol];
      acc = __builtin_amdgcn_wmma_f32_16x16x4_f32(false, afrag, false, bfrag,
                                                  (short)0, acc, false, false);
    }
    __syncthreads();
  }

  // C/D layout: VGPR v, lane -> (M = v + 8*(lane>=16), N = n0 + lcol)
  float* Tb = T + (size_t)b * SEQ * RANK + (size_t)(mt * 16) * RANK;
  const int mhi = (lane >> 4) * 8;
#pragma unroll
  for (int v = 0; v < 8; ++v) {
    Tb[(size_t)(v + mhi) * RANK + n0 + lcol] = acc[v];
  }
}

// ---------------------------------------------------------------------------
// Kernel 2: out[b,i,j] = n[i] + n[j] - 2 * (T[b] @ T[b]^T)[i,j]
// grid = (16 batches, 4 j-groups), block = 128 threads (4 waves, one j-tile each).
// Whole T[b] (256x128 fp32, padded to 132) staged in LDS; norms computed in LDS.
// ---------------------------------------------------------------------------
__global__ __launch_bounds__(128) void gram_wmma(const float* __restrict__ T,
                                                 float* __restrict__ out) {
  __shared__ float ldsT[SEQ * LDST_STR];  // 135168 B
  __shared__ float ldsN[SEQ];             // row norms

  const int b     = blockIdx.x;
  const int jgrp  = blockIdx.y;           // 0..3
  const int t     = threadIdx.x;          // 0..127
  const int wave  = t >> 5;               // 0..3
  const int lane  = t & 31;
  const int lcol  = lane & 15;
  const int khalf = (lane >> 4) << 1;
  const int jt    = jgrp * 4 + wave;      // this wave's j-tile (0..15)

  const float* Tb = T + (size_t)b * SEQ * RANK;

  // ---- stage T[b]: 32768 floats, 64 float4 per thread, padded rows
#pragma unroll 4
  for (int c = 0; c < 64; ++c) {
    const int idx = (c * 128 + t) * 4;
    const int row = idx >> 7;             // /128
    const int col = idx & 127;
    *reinterpret_cast<float4*>(&ldsT[row * LDST_STR + col]) =
        *reinterpret_cast<const float4*>(Tb + idx);
  }
  __syncthreads();

  // ---- row norms: 2 rows per thread
#pragma unroll
  for (int r = 0; r < 2; ++r) {
    const int row = t + r * 128;
    const float* p = &ldsT[row * LDST_STR];
    float s = 0.f;
    for (int k = 0; k < RANK; ++k) s += p[k] * p[k];
    ldsN[row] = s;
  }
  __syncthreads();

  float* outb = out + (size_t)b * SEQ * SEQ;
  const int mhi = (lane >> 4) * 8;
  // B-frag of T*T^T: B[k][n] = T[jt*16+n][k] -> same row, consecutive k per lane
  const float* Bj = &ldsT[(jt * 16 + lcol) * LDST_STR];

  for (int it = 0; it < 16; ++it) {
    const float* Ai = &ldsT[(it * 16 + lcol) * LDST_STR];
    v8f acc = {};
#pragma unroll
    for (int kk = 0; kk < RANK; kk += 4) {
      v2f afrag = *reinterpret_cast<const v2f*>(Ai + kk + khalf);
      v2f bfrag = *reinterpret_cast<const v2f*>(Bj + kk + khalf);
      acc = __builtin_amdgcn_wmma_f32_16x16x4_f32(false, afrag, false, bfrag,
                                                  (short)0, acc, false, false);
    }
#pragma unroll
    for (int v = 0; v < 8; ++v) {
      const int i = it * 16 + v + mhi;
      const int j = jt * 16 + lcol;
      outb[(size_t)i * SEQ + j] = ldsN[i] + ldsN[j] - 2.0f * acc[v];
    }
  }
}

// ---------------------------------------------------------------------------
extern "C" void kernel_launch(void* const* d_in, const int* in_sizes, int n_in,
                              void* d_out, int out_size, void* d_ws, size_t ws_size,
                              hipStream_t stream) {
  const float* batch = (const float*)d_in[0];   // (16, 256, 1024) fp32
  const float* Bm    = (const float*)d_in[1];   // (1024, 128) fp32
  float* out = (float*)d_out;                   // (16, 256, 256) fp32
  float* T   = (float*)d_ws;                    // 16*256*128 fp32 = 2 MB scratch

  proj_wmma<<<dim3(BATCHES, SEQ / 16), 256, 0, stream>>>(batch, Bm, T);
  gram_wmma<<<dim3(BATCHES, 4), 128, 0, stream>>>(T, out);
}